// GraphResnet_7748121002166
// MI455X (gfx1250) — compile-verified
//
#include <hip/hip_runtime.h>
#include <hip/hip_bf16.h>

// GraphResNet (ChebConv K=4 x3 + mix K=2) for gfx1250 / MI455X.
// SpMM (edge gather/scatter): HBM/L2-bound -> float4 gathers + global f32 atomics.
// Dense GEMMs: V_WMMA_F32_16X16X4_F32 (wave32, fp32) with block-shared,
// LDS-staged transposed weights (ds_load_b64 B-fragments, +2 pad = no bank conflicts).

typedef __attribute__((ext_vector_type(2))) float v2f;
typedef __attribute__((ext_vector_type(8))) float v8f;

#define FEAT 64
#define NPAD 66   // LDS row stride (floats) for transposed weights: banks 2n, conflict-free

// ---------------------------------------------------------------- utilities
__global__ void fill_zero_k(float* __restrict__ p, int n) {
  int i = blockIdx.x * blockDim.x + threadIdx.x;
  if (i < n) p[i] = 0.f;
}

__global__ void fill_neg_k(float* __restrict__ o, const float* __restrict__ s, int n) {
  int i = blockIdx.x * blockDim.x + threadIdx.x;
  if (i < n) o[i] = -s[i];
}

__global__ void deg_count_k(const int* __restrict__ src, float* __restrict__ deg, int E) {
  int e = blockIdx.x * blockDim.x + threadIdx.x;
  if (e < E) atomicAdd(&deg[src[e]], 1.0f);
}

__global__ void dinv_k(float* __restrict__ deg, int n) {
  int i = blockIdx.x * blockDim.x + threadIdx.x;
  if (i < n) {
    float d = deg[i];
    deg[i] = (d > 0.f) ? rsqrtf(d) : 0.f;   // counts are integral; max(d,1)==d when d>0
  }
}

__global__ void wedge_k(const int* __restrict__ src, const int* __restrict__ dst,
                        const float* __restrict__ dinv, float* __restrict__ w, int E) {
  int e = blockIdx.x * blockDim.x + threadIdx.x;
  if (e < E) w[e] = -dinv[src[e]] * dinv[dst[e]];
}

// -------------------------------------------------------- SpMM scatter (lmv)
// out[dst[e], :] += scale * w[e] * z[src[e], :]   (64 feats; 16 threads/edge, float4 each)
__global__ void spmm_scatter_k(const int* __restrict__ src, const int* __restrict__ dst,
                               const float* __restrict__ w, const float* __restrict__ z,
                               float* __restrict__ out, float scale, int E) {
  int idx = blockIdx.x * blockDim.x + threadIdx.x;
  int e = idx >> 4;
  if (e >= E) return;
  int f = (idx & 15) << 2;
  int s = src[e], d = dst[e];
  float we = w[e] * scale;
  const float4 v = *reinterpret_cast<const float4*>(z + (size_t)s * FEAT + f);
  float* o = out + (size_t)d * FEAT + f;
  atomicAdd(o + 0, we * v.x);
  atomicAdd(o + 1, we * v.y);
  atomicAdd(o + 2, we * v.z);
  atomicAdd(o + 3, we * v.w);
}

// ------------------------------------------------------------- WMMA helper
__device__ __forceinline__ v8f wmma4(v2f a, v2f b, v8f acc) {
  return __builtin_amdgcn_wmma_f32_16x16x4_f32(false, a, false, b,
                                               (short)0, acc, false, false);
}

// ------------------------------------------------- fused layer GEMM (F=64 out)
// out = relu(h@W0 + t1@W1 + t2@W2 + t3@W3 + kb) + h@Sw + sb
// LDS: 5 weight matrices transposed to [t][n*NPAD + k]  (5*64*66 floats = 84480 B)
__global__ void __launch_bounds__(256)
gemm_layer_k(const float* __restrict__ h, const float* __restrict__ t1,
             const float* __restrict__ t2, const float* __restrict__ t3,
             const float* __restrict__ Wk, const float* __restrict__ kb,
             const float* __restrict__ Sw, const float* __restrict__ sb,
             float* __restrict__ out, int ntiles) {
  extern __shared__ float lds[];

  // ---- stage transposed weights (all waves participate; barrier before any exit)
  for (int i = threadIdx.x; i < 5 * 64 * 64; i += 256) {
    int t = i >> 12;            // matrix 0..4
    int r = i & 4095;           // element within 64x64, k-major
    int k = r >> 6;
    int n = r & 63;
    const float* Wsrc = (t < 4) ? (Wk + (size_t)t * 4096) : Sw;
    lds[t * 64 * NPAD + n * NPAD + k] = Wsrc[r];
  }
  __syncthreads();

  int lane = threadIdx.x & 31;
  int wave = threadIdx.x >> 5;
  int tile = blockIdx.x * 8 + wave;
  int l = lane & 15, half = lane >> 4;
  int mt = (tile < ntiles) ? tile : 0;          // clamp: no early return past barrier
  int m = mt * 16 + l;

  const float* terms[4] = {h, t1, t2, t3};
  v8f zero = {0.f, 0.f, 0.f, 0.f, 0.f, 0.f, 0.f, 0.f};
  v8f accC[4] = {zero, zero, zero, zero};
  v8f accS[4] = {zero, zero, zero, zero};

#pragma unroll
  for (int k0 = 0; k0 < 64; k0 += 4) {
    int k = k0 + 2 * half;
    v2f a[4];
#pragma unroll
    for (int t = 0; t < 4; ++t)
      a[t] = *reinterpret_cast<const v2f*>(terms[t] + (size_t)m * FEAT + k);
#pragma unroll
    for (int nt = 0; nt < 4; ++nt) {
      const float* bbase = lds + (nt * 16 + l) * NPAD + k;
#pragma unroll
      for (int t = 0; t < 4; ++t) {
        v2f b = *reinterpret_cast<const v2f*>(bbase + t * 64 * NPAD);
        accC[nt] = wmma4(a[t], b, accC[nt]);
      }
      v2f bs = *reinterpret_cast<const v2f*>(bbase + 4 * 64 * NPAD);
      accS[nt] = wmma4(a[0], bs, accS[nt]);   // skip term shares A with Cheb T0 (= h)
    }
  }

  if (tile < ntiles) {
#pragma unroll
    for (int nt = 0; nt < 4; ++nt) {
      int ncol = nt * 16 + l;
      float bc = kb[ncol];
      float bsk = sb[ncol];
      size_t base = (size_t)(tile * 16 + 8 * half) * FEAT + ncol;
#pragma unroll
      for (int v = 0; v < 8; ++v)
        out[base + (size_t)v * FEAT] = fmaxf(accC[nt][v] + bc, 0.f) + accS[nt][v] + bsk;
    }
  }
}

// --------------------------------------------- final mix GEMM (concat, 32 out)
// out = h@Wm0[0:64] + x@Wm0[64:128] + t1h@Wm1[0:64] + t1x@Wm1[64:128] + mb
// LDS: 4 (64x32) weight blocks transposed to [t][n*NPAD + k] (4*32*66 floats = 33792 B)
__global__ void __launch_bounds__(256)
gemm_mix_k(const float* __restrict__ h, const float* __restrict__ x,
           const float* __restrict__ t1h, const float* __restrict__ t1x,
           const float* __restrict__ mw, const float* __restrict__ mb,
           float* __restrict__ out, int ntiles) {
  extern __shared__ float lds[];

  const float* Ws[4] = {mw, mw + 64 * 32, mw + 128 * 32, mw + 128 * 32 + 64 * 32};
  for (int i = threadIdx.x; i < 4 * 64 * 32; i += 256) {
    int t = i >> 11;            // block 0..3
    int r = i & 2047;           // 64x32, k-major
    int k = r >> 5;
    int n = r & 31;
    lds[t * 32 * NPAD + n * NPAD + k] = Ws[t][r];
  }
  __syncthreads();

  int lane = threadIdx.x & 31;
  int wave = threadIdx.x >> 5;
  int tile = blockIdx.x * 8 + wave;
  int l = lane & 15, half = lane >> 4;
  int mt = (tile < ntiles) ? tile : 0;
  int m = mt * 16 + l;

  const float* Xs[4] = {h, x, t1h, t1x};
  v8f zero = {0.f, 0.f, 0.f, 0.f, 0.f, 0.f, 0.f, 0.f};
  v8f acc[2] = {zero, zero};

#pragma unroll
  for (int k0 = 0; k0 < 64; k0 += 4) {
    int k = k0 + 2 * half;
    v2f a[4];
#pragma unroll
    for (int t = 0; t < 4; ++t)
      a[t] = *reinterpret_cast<const v2f*>(Xs[t] + (size_t)m * FEAT + k);
#pragma unroll
    for (int nt = 0; nt < 2; ++nt) {
      const float* bbase = lds + (nt * 16 + l) * NPAD + k;
#pragma unroll
      for (int t = 0; t < 4; ++t) {
        v2f b = *reinterpret_cast<const v2f*>(bbase + t * 32 * NPAD);
        acc[nt] = wmma4(a[t], b, acc[nt]);
      }
    }
  }

  if (tile < ntiles) {
#pragma unroll
    for (int nt = 0; nt < 2; ++nt) {
      int ncol = nt * 16 + l;
      float bb = mb[ncol];
      size_t base = (size_t)(tile * 16 + 8 * half) * 32 + ncol;
#pragma unroll
      for (int v = 0; v < 8; ++v)
        out[base + (size_t)v * 32] = acc[nt][v] + bb;
    }
  }
}

// ------------------------------------------------------------------- driver
extern "C" void kernel_launch(void* const* d_in, const int* in_sizes, int n_in,
                              void* d_out, int out_size, void* d_ws, size_t ws_size,
                              hipStream_t stream) {
  const float* x  = (const float*)d_in[0];
  const int*   ei = (const int*)d_in[1];
  const float* kw = (const float*)d_in[2];   // [3,4,64,64]
  const float* kb = (const float*)d_in[3];   // [3,64]
  const float* sw = (const float*)d_in[4];   // [3,64,64]
  const float* sb = (const float*)d_in[5];   // [3,64]
  const float* mw = (const float*)d_in[6];   // [2,128,32]
  const float* mb = (const float*)d_in[7];   // [32]

  const int N = in_sizes[0] / FEAT;          // 100000 (multiple of 16)
  const int E = in_sizes[1] / 2;             // 1200000
  const int* src = ei;
  const int* dst = ei + E;

  float* ws    = (float*)d_ws;
  float* deg   = ws;                          // N   (becomes dinv)
  float* wedge = deg + N;                     // E
  float* bufA  = wedge + E;                   // N*64
  float* bufB  = bufA + (size_t)N * FEAT;     // N*64
  float* t1    = bufB + (size_t)N * FEAT;     // N*64
  float* t2    = t1 + (size_t)N * FEAT;       // N*64
  float* t3    = t2 + (size_t)N * FEAT;       // N*64

  const int TB = 256;
  const int nf = N * FEAT;
  const int gN  = (N + TB - 1) / TB;
  const int gE  = (E + TB - 1) / TB;
  const int gNF = (nf + TB - 1) / TB;
  const int gSC = (E * 16 + TB - 1) / TB;     // 16 threads / edge
  const int ntiles = N / 16;
  const int gGE = (ntiles + 7) / 8;           // 8 waves / block
  const size_t ldsLayer = (size_t)5 * 64 * NPAD * sizeof(float);  // 84480 B
  const size_t ldsMix   = (size_t)4 * 32 * NPAD * sizeof(float);  // 33792 B

  // edge weights of scaled Laplacian
  fill_zero_k<<<gN, TB, 0, stream>>>(deg, N);
  deg_count_k<<<gE, TB, 0, stream>>>(src, deg, E);
  dinv_k<<<gN, TB, 0, stream>>>(deg, N);
  wedge_k<<<gE, TB, 0, stream>>>(src, dst, deg, wedge, E);

  // 3 ChebConv(K=4) + skip layers, ping-pong h through bufA/bufB
  const float* hcur = x;
  float* outs[3] = {bufA, bufB, bufA};
  for (int i = 0; i < 3; ++i) {
    // T1 = lmv(T0)
    fill_zero_k<<<gNF, TB, 0, stream>>>(t1, nf);
    spmm_scatter_k<<<gSC, TB, 0, stream>>>(src, dst, wedge, hcur, t1, 1.0f, E);
    // T2 = 2*lmv(T1) - T0
    fill_neg_k<<<gNF, TB, 0, stream>>>(t2, hcur, nf);
    spmm_scatter_k<<<gSC, TB, 0, stream>>>(src, dst, wedge, t1, t2, 2.0f, E);
    // T3 = 2*lmv(T2) - T1
    fill_neg_k<<<gNF, TB, 0, stream>>>(t3, t1, nf);
    spmm_scatter_k<<<gSC, TB, 0, stream>>>(src, dst, wedge, t2, t3, 2.0f, E);
    // h = relu(sum Tk@Wk + kb) + h@Sw + sb
    gemm_layer_k<<<gGE, TB, ldsLayer, stream>>>(hcur, t1, t2, t3,
                                                kw + (size_t)i * 4 * 64 * 64, kb + i * 64,
                                                sw + (size_t)i * 64 * 64, sb + i * 64,
                                                outs[i], ntiles);
    hcur = outs[i];
  }

  // mix conv on concat(h, x): t1 = lmv(h), t2 = lmv(x)
  fill_zero_k<<<gNF, TB, 0, stream>>>(t1, nf);
  spmm_scatter_k<<<gSC, TB, 0, stream>>>(src, dst, wedge, hcur, t1, 1.0f, E);
  fill_zero_k<<<gNF, TB, 0, stream>>>(t2, nf);
  spmm_scatter_k<<<gSC, TB, 0, stream>>>(src, dst, wedge, x, t2, 1.0f, E);
  gemm_mix_k<<<gGE, TB, ldsMix, stream>>>(hcur, x, t1, t2, mw, mb, (float*)d_out, ntiles);
}